// QKCodebook_68315749810229
// MI455X (gfx1250) — compile-verified
//
#include <hip/hip_runtime.h>
#include <hip/hip_bf16.h>

typedef __attribute__((ext_vector_type(2))) float v2f;
typedef __attribute__((ext_vector_type(8))) float v8f;

#define TOK_TILE   16
#define ENC_STRIDE 516   // 512 + 4 pad: 16 rows map to 16 distinct LDS bank groups

// ---------------------------------------------------------------------------
// Weight re-layout: W(k, c) [512 x 65536]  ->  W'(k/2, c, k%2)
// so that a lane's B-fragment (rows k+2hi, k+2hi+1 at column c) is one b64.
// ---------------------------------------------------------------------------
__global__ void relayout_pairs(const float* __restrict__ w1,
                               const float* __restrict__ w2,
                               float* __restrict__ o1,
                               float* __restrict__ o2) {
  const float* in  = blockIdx.z ? w2 : w1;
  float*       out = blockIdx.z ? o2 : o1;
  const int c = blockIdx.x * 256 + threadIdx.x;   // 0..65535
  const int p = blockIdx.y;                       // k-pair 0..255
  const float a = in[(size_t)(2 * p)     * 65536u + c];
  const float b = in[(size_t)(2 * p + 1) * 65536u + c];
  float2 v; v.x = a; v.y = b;
  *(float2*)(out + (size_t)p * 131072u + 2u * (unsigned)c) = v;  // 8B/lane contig
}

// ---------------------------------------------------------------------------
// One phase of the dynamic MLP:
//   acc[t,m,h] = sum_g  sin[t,m,g] * ( sum_k enc[t,k] * W[k,g,h] )
// Wave owns h-tile [h0, h0+16); tokens striped per WMMA C layout.
// TR=true : W is pair-interleaved  -> 1 global_load_b64 per WMMA
// TR=false: W is reference layout  -> 2 global_load_b32 per WMMA
// ---------------------------------------------------------------------------
template <bool TR>
__device__ __forceinline__ void mlp_phase(
    const float* __restrict__ W,
    const float* enc_lds,             // [16][ENC_STRIDE]
    const float* sin_lds,             // [(t*2+m)*256 + g]
    int lo, int hi, int h0,
    float* acc)                       // [16] = acc[i*2+m], t = i + 8*hi
{
#pragma unroll
  for (int j = 0; j < 16; ++j) acc[j] = 0.f;
  const float* encp = enc_lds + lo * ENC_STRIDE + 2 * hi;

  for (int g = 0; g < 256; g += 2) {
    v8f c0 = {0,0,0,0,0,0,0,0};
    v8f c1 = {0,0,0,0,0,0,0,0};
    const int c = g * 256 + h0 + lo;              // B column for this lane
    const float* pb = TR
        ? W + (size_t)hi * 131072u + 2u * (unsigned)c          // p = k/2 + hi
        : W + (size_t)(2 * hi) * 65536u + (unsigned)c;         // row k + 2hi
#pragma unroll 8
    for (int k = 0; k < 512; k += 4) {
      v2f a = *(const v2f*)(encp + k);            // ds_load_b64: K rows k+2hi,+1
      v2f b0, b1;
      if (TR) {
        b0 = *(const v2f*)(pb);                   // global_load_b64 (g)
        b1 = *(const v2f*)(pb + 512);             // global_load_b64 (g+1)
      } else {
        b0[0] = pb[0];        b0[1] = pb[65536];
        b1[0] = pb[256];      b1[1] = pb[65536 + 256];
      }
      c0 = __builtin_amdgcn_wmma_f32_16x16x4_f32(false, a, false, b0, (short)0, c0, false, false);
      c1 = __builtin_amdgcn_wmma_f32_16x16x4_f32(false, a, false, b1, (short)0, c1, false, false);
      pb += TR ? (size_t)2 * 131072 : (size_t)4 * 65536;       // advance 4 k
    }
    // consume the two C tiles immediately (fusion: C never leaves registers)
#pragma unroll
    for (int i = 0; i < 8; ++i) {
      const int t = i + 8 * hi;
#pragma unroll
      for (int m = 0; m < 2; ++m) {
        acc[2 * i + m] += sin_lds[(t * 2 + m) * 256 + g]     * c0[i]
                        + sin_lds[(t * 2 + m) * 256 + g + 1] * c1[i];
      }
    }
  }
}

template <bool TR>
__global__ __launch_bounds__(512, 1) void qk_codebook_fused(
    const float* __restrict__ enc,   // (2048, 512)
    const float* __restrict__ muE,   // (512, 256)
    const float* __restrict__ sgE,   // (512, 256)
    const float* __restrict__ w1,    // (512,256,256) or pair-interleaved
    const float* __restrict__ fb1,   // (512, 256)
    const float* __restrict__ w2,
    const float* __restrict__ fb2,
    const float* __restrict__ eps,   // (32, 2, 64, 256)
    float* __restrict__ out)         // (64, 64, 256)
{
  __shared__ float enc_lds[TOK_TILE * ENC_STRIDE];  // 33.0 KB
  __shared__ float s_in [32 * 256];                 // 32 KB  reparam slots
  __shared__ float x_mid[32 * 256];                 // 32 KB  layer-1 activations
  __shared__ float b1_lds[16 * 256];                // 16 KB  enc@fc1_b
  __shared__ float b2_lds[16 * 256];                // 16 KB  enc@fc2_b

  const int tid = threadIdx.x;
  const int T0  = blockIdx.x * TOK_TILE;

  // ---- stage encodings tile (16 x 512) into padded LDS ----
#pragma unroll
  for (int j = 0; j < 16; ++j) {
    const int f = tid + 512 * j;           // 0..8191
    const int t = f >> 9, k = f & 511;
    enc_lds[t * ENC_STRIDE + k] = enc[(size_t)(T0 + t) * 512 + k];
  }
  __syncthreads();

  // ---- prologue: slot params + mixed biases (tiny vs main GEMM) ----
  {
    const int dcol = tid & 255;
    const int tg   = tid >> 8;             // tokens [8*tg, 8*tg+8)
    float aMu[8], aSg[8], aB1[8], aB2[8];
#pragma unroll
    for (int j = 0; j < 8; ++j) { aMu[j] = aSg[j] = aB1[j] = aB2[j] = 0.f; }
    for (int k = 0; k < 512; ++k) {
      const float vMu = muE[k * 256 + dcol];
      const float vSg = sgE[k * 256 + dcol];
      const float vB1 = fb1[k * 256 + dcol];
      const float vB2 = fb2[k * 256 + dcol];
#pragma unroll
      for (int j = 0; j < 8; ++j) {
        const float e = enc_lds[(8 * tg + j) * ENC_STRIDE + k];  // LDS broadcast
        aMu[j] = fmaf(e, vMu, aMu[j]);
        aSg[j] = fmaf(e, vSg, aSg[j]);
        aB1[j] = fmaf(e, vB1, aB1[j]);
        aB2[j] = fmaf(e, vB2, aB2[j]);
      }
    }
#pragma unroll
    for (int j = 0; j < 8; ++j) {
      const int t = 8 * tg + j, T = T0 + t;
      const int bb = T >> 6, nn = T & 63;
      const float sig = __expf(0.5f * aSg[j]);
#pragma unroll
      for (int m = 0; m < 2; ++m) {
        const float ep = eps[((size_t)((bb * 2 + m) * 64 + nn)) * 256 + dcol];
        s_in[(t * 2 + m) * 256 + dcol] = fmaf(sig, ep, aMu[j]);
      }
      b1_lds[t * 256 + dcol] = aB1[j];
      b2_lds[t * 256 + dcol] = aB2[j];
    }
  }
  __syncthreads();

  const int lane = tid & 31;
  const int lo = lane & 15, hi = lane >> 4;
  const int h0 = (tid >> 5) * 16;          // 16 waves -> h tiles 0..255

  float acc[16];

  // ---- layer 1: dynamic W1 formed & consumed tile-by-tile ----
  mlp_phase<TR>(w1, enc_lds, s_in, lo, hi, h0, acc);
#pragma unroll
  for (int i = 0; i < 8; ++i) {
    const int t = i + 8 * hi;
#pragma unroll
    for (int m = 0; m < 2; ++m) {
      const float v = acc[2 * i + m] + b1_lds[t * 256 + h0 + lo];
      x_mid[(t * 2 + m) * 256 + h0 + lo] = fmaxf(v, 0.f);
    }
  }
  __syncthreads();

  // ---- layer 2 ----
  mlp_phase<TR>(w2, enc_lds, x_mid, lo, hi, h0, acc);
#pragma unroll
  for (int i = 0; i < 8; ++i) {
    const int t = i + 8 * hi, T = T0 + t;
    const int bb = T >> 6, nn = T & 63;
#pragma unroll
    for (int m = 0; m < 2; ++m) {
      const float v = acc[2 * i + m] + b2_lds[t * 256 + h0 + lo];
      out[((size_t)((bb * 2 + m) * 64 + nn)) * 256 + h0 + lo] = fmaxf(v, 0.f);
    }
  }
}

extern "C" void kernel_launch(void* const* d_in, const int* in_sizes, int n_in,
                              void* d_out, int out_size, void* d_ws, size_t ws_size,
                              hipStream_t stream) {
  (void)in_sizes; (void)n_in; (void)out_size;
  const float* enc = (const float*)d_in[0];
  const float* muE = (const float*)d_in[1];
  const float* sgE = (const float*)d_in[2];
  const float* w1  = (const float*)d_in[3];
  const float* fb1 = (const float*)d_in[4];
  const float* w2  = (const float*)d_in[5];
  const float* fb2 = (const float*)d_in[6];
  const float* eps = (const float*)d_in[7];
  float* out = (float*)d_out;

  const size_t kWtFloats = (size_t)512 * 256 * 256;          // 33,554,432 per mat
  const bool use_tr = ws_size >= 2u * kWtFloats * sizeof(float);  // 268 MB

  if (use_tr) {
    float* wt1 = (float*)d_ws;
    float* wt2 = wt1 + kWtFloats;
    // one-time (per launch) pair-interleave of both weight tensors
    relayout_pairs<<<dim3(256, 256, 2), 256, 0, stream>>>(w1, w2, wt1, wt2);
    qk_codebook_fused<true><<<128, 512, 0, stream>>>(enc, muE, sgE, wt1, fb1,
                                                     wt2, fb2, eps, out);
  } else {
    qk_codebook_fused<false><<<128, 512, 0, stream>>>(enc, muE, sgE, w1, fb1,
                                                      w2, fb2, eps, out);
  }
}